// TokenSetRouter_88003879895288
// MI455X (gfx1250) — compile-verified
//
#include <hip/hip_runtime.h>
#include <math.h>

typedef __attribute__((ext_vector_type(2))) float v2f;
typedef __attribute__((ext_vector_type(8))) float v8f;
typedef __attribute__((ext_vector_type(4))) int   v4i;

#define D_DIM 1024
#define SMAX  2048

// LDS tile geometry: BM=128, BN=64, BK=32, stride 36 floats.
// Stride 36 => row offsets are distinct multiples of 4 banks (cycle 16), so the
// 32-lane ds_load_b64 fragment reads (low half at k, high half at k+2) are
// bank-conflict free.
#define BK   32
#define LDSS 36

#if defined(__HIPCC__) && __has_builtin(__builtin_amdgcn_global_load_async_to_lds_b128)
#define HAVE_ASYNC_LDS 1
#else
#define HAVE_ASYNC_LDS 0
#endif
#if defined(__HIPCC__) && __has_builtin(__builtin_amdgcn_s_wait_asynccnt)
#define HAVE_WAIT_ASYNC 1
#else
#define HAVE_WAIT_ASYNC 0
#endif

// Copy 16 bytes global -> LDS. Async (GLOBAL_LOAD_ASYNC_TO_LDS_B128, ASYNCcnt)
// when the toolchain exposes the builtin; otherwise stage through VGPRs.
// Builtin signature (from hipcc diagnostic): param0 = v4i* in addrspace(1)
// (printed as "__device__"), non-const; param1 = LDS-side pointer; then
// immediate offset and cpol.
__device__ __forceinline__ void copy16_g2l(const float* g, float* l) {
#if HAVE_ASYNC_LDS
  float* gnc = const_cast<float*>(g);
  __builtin_amdgcn_global_load_async_to_lds_b128(
      (__attribute__((address_space(1))) v4i*)gnc,
      (__attribute__((address_space(3))) v4i*)l,
      0, 0);
#else
  *reinterpret_cast<float4*>(l) = *reinterpret_cast<const float4*>(g);
#endif
}

__device__ __forceinline__ void g2l_wait() {
#if HAVE_ASYNC_LDS
#if HAVE_WAIT_ASYNC
  __builtin_amdgcn_s_wait_asynccnt(0);
#else
  asm volatile("s_wait_asynccnt 0" ::: "memory");
#endif
#endif
}

// Out[M,N] = X[M,K] @ W[N,K]^T + bias[N]
// Block: 256 threads = 8 waves (4 along M x 2 along N); block tile 128x64.
// Double-buffered LDS staging of 128x32 X-tiles and 64x32 W-tiles, async
// global->LDS fill overlapped with V_WMMA_F32_16X16X4_F32 compute.
__global__ __launch_bounds__(256) void gemm_xwT_bias(
    const float* __restrict__ X, const float* __restrict__ W,
    const float* __restrict__ bias, float* __restrict__ Out,
    int M, int N, int K)
{
  __shared__ float Xs[2][128 * LDSS];
  __shared__ float Ws[2][64 * LDSS];

  const int tid  = threadIdx.x;
  const int lane = tid & 31;   // wave32
  const int wave = tid >> 5;   // 0..7
  const int half = lane >> 4;  // 0 or 1 -> owns K pair (k, k+1) or (k+2, k+3)
  const int mr   = lane & 15;

  const int blockM = blockIdx.x * 128;
  const int blockN = blockIdx.y * 64;
  const int wm0 = (wave & 3) * 32;   // wave tile origin inside block (M)
  const int wn0 = (wave >> 2) * 32;  // wave tile origin inside block (N)

  v8f c00 = {}; v8f c01 = {}; v8f c10 = {}; v8f c11 = {};

  // Cooperative staging of one K-chunk into LDS buffer `buf`.
  auto stage = [&](int buf, int kbase) {
    // X tile: 128 rows x 32 floats = 1024 float4 -> 4 per thread, coalesced rows.
    #pragma unroll
    for (int i = 0; i < 4; ++i) {
      const int idx = tid + i * 256;
      const int row = idx >> 3;
      const int c   = (idx & 7) * 4;
      copy16_g2l(X + (size_t)(blockM + row) * K + kbase + c,
                 &Xs[buf][row * LDSS + c]);
    }
    // W tile: 64 rows x 32 floats = 512 float4 -> 2 per thread.
    #pragma unroll
    for (int i = 0; i < 2; ++i) {
      const int idx = tid + i * 256;
      const int row = idx >> 3;
      const int c   = (idx & 7) * 4;
      copy16_g2l(W + (size_t)(blockN + row) * K + kbase + c,
                 &Ws[buf][row * LDSS + c]);
    }
  };

  stage(0, 0);
  g2l_wait();
  __syncthreads();

  int buf = 0;
  for (int kc = 0; kc < K; kc += BK) {
    if (kc + BK < K) stage(buf ^ 1, kc + BK);  // prefetch next chunk (async)

    const float* xs0 = &Xs[buf][(wm0 + mr) * LDSS + 2 * half];
    const float* xs1 = &Xs[buf][(wm0 + 16 + mr) * LDSS + 2 * half];
    const float* ws0 = &Ws[buf][(wn0 + mr) * LDSS + 2 * half];
    const float* ws1 = &Ws[buf][(wn0 + 16 + mr) * LDSS + 2 * half];

    #pragma unroll
    for (int k = 0; k < BK; k += 4) {
      v2f a0 = *reinterpret_cast<const v2f*>(xs0 + k);
      v2f a1 = *reinterpret_cast<const v2f*>(xs1 + k);
      v2f b0 = *reinterpret_cast<const v2f*>(ws0 + k);
      v2f b1 = *reinterpret_cast<const v2f*>(ws1 + k);
      // (neg_a, A, neg_b, B, c_mod, C, reuse_a, reuse_b)
      c00 = __builtin_amdgcn_wmma_f32_16x16x4_f32(false, a0, false, b0, (short)0, c00, false, false);
      c01 = __builtin_amdgcn_wmma_f32_16x16x4_f32(false, a0, false, b1, (short)0, c01, false, false);
      c10 = __builtin_amdgcn_wmma_f32_16x16x4_f32(false, a1, false, b0, (short)0, c10, false, false);
      c11 = __builtin_amdgcn_wmma_f32_16x16x4_f32(false, a1, false, b1, (short)0, c11, false, false);
    }

    g2l_wait();        // next chunk resident in LDS
    __syncthreads();   // all waves done reading `buf`, fills visible
    buf ^= 1;
  }

  // C/D layout: VGPR r holds row (r + 8*half), col = lane&15 (ISA 7.12.2)
  const int col0 = blockN + wn0 + mr;
  const int col1 = col0 + 16;
  const float bias0 = bias[col0];
  const float bias1 = bias[col1];
  const int row0 = blockM + wm0 + 8 * half;
  #pragma unroll
  for (int r = 0; r < 8; ++r) {
    Out[(size_t)(row0 + r)      * N + col0] = c00[r] + bias0;
    Out[(size_t)(row0 + r)      * N + col1] = c01[r] + bias1;
    Out[(size_t)(row0 + 16 + r) * N + col0] = c10[r] + bias0;
    Out[(size_t)(row0 + 16 + r) * N + col1] = c11[r] + bias1;
  }
}

// One block (256 threads) per token: logits vs this batch's descriptor slab,
// top-8 (lowest-index tie-break, matching lax.top_k), softmax over the 8,
// then mix[token,:] = sum_j gate[j] * Z[topidx[j], :].
__global__ __launch_bounds__(256) void router_topk_mix(
    const float* __restrict__ Tproj, const float* __restrict__ Dall,
    const float* __restrict__ Z, const int* __restrict__ q_ptrs,
    float* __restrict__ mix, int L)
{
  __shared__ float t[D_DIM];
  __shared__ float logits[SMAX];
  __shared__ float redv[256];
  __shared__ int   redi[256];
  __shared__ float topv[8];
  __shared__ int   topi[8];
  __shared__ float gate[8];

  const int tid   = threadIdx.x;
  const int token = blockIdx.x;
  const int b     = token / L;
  const int start = q_ptrs[b];
  int len = q_ptrs[b + 1] - start;
  if (len > SMAX) len = SMAX;

  reinterpret_cast<float4*>(t)[tid] =
      reinterpret_cast<const float4*>(Tproj + (size_t)token * D_DIM)[tid];
  __syncthreads();

  const float4* t4 = reinterpret_cast<const float4*>(t);
  for (int s = tid; s < len; s += 256) {
    const float4* drow = reinterpret_cast<const float4*>(Dall + (size_t)(start + s) * D_DIM);
    float ax = 0.f, ay = 0.f, az = 0.f, aw = 0.f;
    #pragma unroll 8
    for (int kk = 0; kk < D_DIM / 4; ++kk) {
      float4 tv = t4[kk];
      float4 dv = drow[kk];
      ax = fmaf(tv.x, dv.x, ax);
      ay = fmaf(tv.y, dv.y, ay);
      az = fmaf(tv.z, dv.z, az);
      aw = fmaf(tv.w, dv.w, aw);
    }
    logits[s] = (ax + ay) + (az + aw);
  }
  __syncthreads();

  const float NEG = -__builtin_huge_valf();
  for (int it = 0; it < 8; ++it) {
    float bv = NEG; int bi = 0x7FFFFFFF;
    for (int s = tid; s < len; s += 256) {
      float v = logits[s];
      if (v > bv || (v == bv && s < bi)) { bv = v; bi = s; }
    }
    redv[tid] = bv; redi[tid] = bi;
    __syncthreads();
    for (int off = 128; off > 0; off >>= 1) {
      if (tid < off) {
        float v2 = redv[tid + off]; int i2 = redi[tid + off];
        if (v2 > redv[tid] || (v2 == redv[tid] && i2 < redi[tid])) {
          redv[tid] = v2; redi[tid] = i2;
        }
      }
      __syncthreads();
    }
    if (tid == 0) {
      topv[it] = redv[0];
      topi[it] = start + redi[0];
      logits[redi[0]] = NEG;
    }
    __syncthreads();
  }

  if (tid == 0) {
    float m = topv[0];
    for (int j = 1; j < 8; ++j) m = fmaxf(m, topv[j]);
    float e[8];
    float sum = 0.f;
    for (int j = 0; j < 8; ++j) { e[j] = expf(topv[j] - m); sum += e[j]; }
    float inv = 1.f / sum;
    for (int j = 0; j < 8; ++j) gate[j] = e[j] * inv;
  }
  __syncthreads();

  float ax = 0.f, ay = 0.f, az = 0.f, aw = 0.f;
  #pragma unroll
  for (int j = 0; j < 8; ++j) {
    const float4 z = reinterpret_cast<const float4*>(Z + (size_t)topi[j] * D_DIM)[tid];
    const float gj = gate[j];
    ax = fmaf(gj, z.x, ax);
    ay = fmaf(gj, z.y, ay);
    az = fmaf(gj, z.z, az);
    aw = fmaf(gj, z.w, aw);
  }
  float4 o; o.x = ax; o.y = ay; o.z = az; o.w = aw;
  reinterpret_cast<float4*>(mix + (size_t)token * D_DIM)[tid] = o;
}

extern "C" void kernel_launch(void* const* d_in, const int* in_sizes, int n_in,
                              void* d_out, int out_size, void* d_ws, size_t ws_size,
                              hipStream_t stream)
{
  const float* token_states = (const float*)d_in[0];  // (4,2048,1024)
  const float* Z_sets       = (const float*)d_in[1];  // (2048,16,64) == (2048,1024)
  const float* desc_q       = (const float*)d_in[2];  // (2048,1024)
  const int*   q_ptrs       = (const int*)d_in[3];    // (5,)
  const float* Wg_w         = (const float*)d_in[4];
  const float* Wg_b         = (const float*)d_in[5];
  const float* Wd_w         = (const float*)d_in[6];
  const float* Wd_b         = (const float*)d_in[7];
  const float* out_w        = (const float*)d_in[8];
  const float* out_b        = (const float*)d_in[9];

  const int Bv = 4, Lv = 2048, Dv = 1024, Nv = 2048;
  const int BL = Bv * Lv;  // 8192

  // Workspace layout: Tproj (32MB) | Dall (8MB) | mix (32MB) => 72MB total
  char* ws = (char*)d_ws;
  float* Tproj = (float*)ws;
  float* Dall  = (float*)(ws + (size_t)BL * Dv * sizeof(float));
  float* mixb  = (float*)(ws + (size_t)BL * Dv * sizeof(float)
                             + (size_t)Nv * Dv * sizeof(float));

  dim3 blk(256);

  gemm_xwT_bias<<<dim3(BL / 128, Dv / 64), blk, 0, stream>>>(
      token_states, Wg_w, Wg_b, Tproj, BL, Dv, Dv);

  gemm_xwT_bias<<<dim3(Nv / 128, Dv / 64), blk, 0, stream>>>(
      desc_q, Wd_w, Wd_b, Dall, Nv, Dv, Dv);

  router_topk_mix<<<dim3(BL), blk, 0, stream>>>(
      Tproj, Dall, Z_sets, q_ptrs, mixb, Lv);

  gemm_xwT_bias<<<dim3(BL / 128, Dv / 64), blk, 0, stream>>>(
      mixb, out_w, out_b, (float*)d_out, BL, Dv, Dv);
}